// Model_25786983645381
// MI455X (gfx1250) — compile-verified
//
#include <hip/hip_runtime.h>

#define B_   8
#define T_   2048
#define DV_  1024
#define DA_  128
#define D1_  512
#define D2_  128
#define DH_  257
#define DHP_ 260     // DH padded to multiple of 4 (WMMA K step)
#define PXS_ 272     // proj_x row stride (16-aligned)
#define DG_  128
#define NEG_ 0.01f

#define KC_   32     // K-chunk staged in LDS
#define BPS_  160    // LDS k-pair row stride, 64-wide panel  (160%64==32)
#define GPS_  288    // LDS k-pair row stride, 128-wide panel (288%64==32)

typedef __attribute__((ext_vector_type(2))) float v2f;
typedef __attribute__((ext_vector_type(8))) float v8f;

__device__ __forceinline__ v8f wmma4(v2f a, v2f b, v8f c) {
  // D = A(16x4 f32) * B(4x16 f32) + C(16x16 f32)  -> v_wmma_f32_16x16x4_f32
  return __builtin_amdgcn_wmma_f32_16x16x4_f32(false, a, false, b, (short)0, c,
                                               false, false);
}
__device__ __forceinline__ float leaky_(float x) { return x >= 0.f ? x : NEG_ * x; }

__device__ __forceinline__ void interleave2(const float4& x, const float4& y,
                                            float4& lo, float4& hi) {
  lo.x = x.x; lo.y = y.x; lo.z = x.y; lo.w = y.y;
  hi.x = x.z; hi.y = y.z; hi.z = x.w; hi.w = y.w;
}

// ---------------- weight prep ----------------
__global__ void k_transpose(const float* __restrict__ src, float* __restrict__ dst,
                            int rows, int cols) {
  int idx = blockIdx.x * 256 + threadIdx.x;
  if (idx < rows * cols) {
    int r = idx / cols, c = idx % cols;
    dst[c * rows + r] = src[idx];
  }
}

__global__ void k_padcopy(const float* __restrict__ src, float* __restrict__ dst,
                          int rows, int cols, int prows) {
  int idx = blockIdx.x * 256 + threadIdx.x;
  if (idx < prows * cols) {
    int r = idx / cols, c = idx % cols;
    dst[idx] = (r < rows) ? src[r * cols + c] : 0.f;
  }
}

// ------- WMMA fp32 GEMM, LDS double-buffered B panel: C = act(A*B + bias) ----
// A: M x K (row stride lda), B: K x N (row stride ldb). Block = 4 waves, each
// wave owns a 16x64 tile; B panel (KCx64) shared via LDS, k-pair interleaved
// ([k/2][n][2]) so each B fragment is one aligned ds_load_b64.
__global__ __launch_bounds__(128) void k_gemm(
    const float* __restrict__ A, int lda, const float* __restrict__ Bm, int ldb,
    const float* __restrict__ bias, float* __restrict__ C, int ldc, int K,
    int doleaky) {
  __shared__ __align__(16) float Bsh[2][16 * BPS_];
  const int tid = threadIdx.x;
  const int lane = tid & 31, wv = tid >> 5;
  const int half = lane >> 4, l16 = lane & 15;
  const int r0 = blockIdx.y * 64 + wv * 16;
  const int n0 = blockIdx.x * 64;

  const float* ap = A + (size_t)(r0 + l16) * lda + 2 * half;
  const float* bp = Bm + n0;

  v8f acc[4] = {};
  float4 bx[2], by[2];
  v2f a0[8], a1[8];

  const int nchunk = K / KC_;
  if (nchunk > 0) {  // preload + stage chunk 0
#pragma unroll
    for (int q = 0; q < 2; ++q) {
      int u = tid + 128 * q, kp = u >> 4, n4 = (u & 15) * 4;
      bx[q] = *(const float4*)(bp + (size_t)(2 * kp) * ldb + n4);
      by[q] = *(const float4*)(bp + (size_t)(2 * kp + 1) * ldb + n4);
    }
#pragma unroll
    for (int s = 0; s < 8; ++s) a0[s] = *(const v2f*)(ap + 4 * s);
#pragma unroll
    for (int q = 0; q < 2; ++q) {
      int u = tid + 128 * q, kp = u >> 4, n4 = (u & 15) * 4;
      float4 lo, hi;
      interleave2(bx[q], by[q], lo, hi);
      *(float4*)&Bsh[0][kp * BPS_ + n4 * 2] = lo;
      *(float4*)&Bsh[0][kp * BPS_ + n4 * 2 + 4] = hi;
    }
  }
  __syncthreads();

  for (int c = 0; c < nchunk; ++c) {
    const int cur = c & 1;
    const int k0n = (c + 1) * KC_;
    if (c + 1 < nchunk) {  // prefetch next chunk (loads issue, no wait yet)
#pragma unroll
      for (int q = 0; q < 2; ++q) {
        int u = tid + 128 * q, kp = u >> 4, n4 = (u & 15) * 4;
        bx[q] = *(const float4*)(bp + (size_t)(k0n + 2 * kp) * ldb + n4);
        by[q] = *(const float4*)(bp + (size_t)(k0n + 2 * kp + 1) * ldb + n4);
      }
#pragma unroll
      for (int s = 0; s < 8; ++s) a1[s] = *(const v2f*)(ap + k0n + 4 * s);
    }
#pragma unroll
    for (int ks = 0; ks < 8; ++ks) {  // 32 WMMAs hide the prefetch latency
      v2f a = a0[ks];
      const float* bl = &Bsh[cur][(ks * 2 + half) * BPS_ + l16 * 2];
#pragma unroll
      for (int t = 0; t < 4; ++t) {
        v2f b = *(const v2f*)(bl + t * 32);  // one ds_load_b64 per fragment
        acc[t] = wmma4(a, b, acc[t]);
      }
    }
    if (c + 1 < nchunk) {
#pragma unroll
      for (int q = 0; q < 2; ++q) {
        int u = tid + 128 * q, kp = u >> 4, n4 = (u & 15) * 4;
        float4 lo, hi;
        interleave2(bx[q], by[q], lo, hi);
        *(float4*)&Bsh[cur ^ 1][kp * BPS_ + n4 * 2] = lo;
        *(float4*)&Bsh[cur ^ 1][kp * BPS_ + n4 * 2 + 4] = hi;
      }
#pragma unroll
      for (int s = 0; s < 8; ++s) a0[s] = a1[s];  // reg ping-pong, no scratch
    }
    __syncthreads();
  }
  for (int k0 = nchunk * KC_; k0 < K; k0 += 4) {  // K%32 tail (K=260 case)
    v2f a = *(const v2f*)(ap + k0);
    const float* bg = Bm + (size_t)(k0 + 2 * half) * ldb + n0 + l16;
#pragma unroll
    for (int t = 0; t < 4; ++t) {
      v2f b;
      b.x = bg[t * 16];
      b.y = bg[ldb + t * 16];
      acc[t] = wmma4(a, b, acc[t]);
    }
  }
#pragma unroll
  for (int t = 0; t < 4; ++t) {
    int n = n0 + t * 16 + l16;
    float bv = bias ? bias[n] : 0.f;
#pragma unroll
    for (int v = 0; v < 8; ++v) {
      int r = r0 + v + 8 * half;
      float val = acc[t][v] + bv;
      if (doleaky) val = leaky_(val);
      C[(size_t)r * ldc + n] = val;
    }
  }
}

// ---------------- Lorentz expmap0: build proj_x (px) and signed copy (sx) ----
__global__ __launch_bounds__(256) void k_proj(const float* __restrict__ xv2,
                                              const float* __restrict__ inputs,
                                              float* __restrict__ px,
                                              float* __restrict__ sx) {
  int row = blockIdx.x * 8 + (threadIdx.x >> 5);  // 0..16383 == b*T+t
  int lane = threadIdx.x & 31;
  const float* xa = inputs + (size_t)row * (DV_ + DA_) + DV_;
  const float* xv = xv2 + (size_t)row * D2_;
  float vals[8];
  float ss = 0.f;
#pragma unroll
  for (int q = 0; q < 8; ++q) {
    int d = lane + 32 * q;  // 0..255
    float v = (d < D2_) ? xv[d] : xa[d - D2_];
    vals[q] = v;
    ss += v * v;
  }
#pragma unroll
  for (int o = 16; o > 0; o >>= 1) ss += __shfl_xor(ss, o, 32);
  float n = sqrtf(fmaxf(ss, 1e-12f));
  float sh = sinhf(n) / n;
  float* prow = px + (size_t)row * PXS_;
  float* srow = sx + (size_t)row * PXS_;
#pragma unroll
  for (int q = 0; q < 8; ++q) {
    int d = lane + 32 * q;
    float pv = sh * vals[q];
    prow[1 + d] = pv;
    srow[1 + d] = -pv;  // Minkowski sign (+,-,-,...) folded into A side
  }
  if (lane == 0) {
    float c = coshf(n);
    prow[0] = c;
    srow[0] = c;
  }
  if (lane < PXS_ - DH_) {  // zero pad 257..271
    prow[DH_ + lane] = 0.f;
    srow[DH_ + lane] = 0.f;
  }
}

// ------- adj softmax-attention fused, fully LDS-fed:  x1 = leaky(adj @ g1) ---
// dynamic LDS layout (floats):
//   AL  : 4*16*272  per-wave sx row tiles
//   BJ  : 32*272    px rows of current 32-col j-chunk (shared by 4 waves)
//   G1L : 16*288    g1 rows of current j-chunk, k-pair interleaved
//   WL  : 4*256     per-wave 16x16 softmax-weight tile
//   DL  : 64        per-wave row denominators
__global__ __launch_bounds__(128) void k_adjattn(
    const float* __restrict__ sx, const float* __restrict__ px,
    const float* __restrict__ g1, const int* __restrict__ seqlen,
    float* __restrict__ outx) {
  extern __shared__ __align__(16) float smem[];
  float* AL = smem;                  // 17408
  float* BJ = AL + 4 * 16 * PXS_;    // 8704
  float* G1L = BJ + 32 * PXS_;       // 4608
  float* WL = G1L + 16 * GPS_;       // 1024
  float* DL = WL + 4 * 256;          // 64

  const int tid = threadIdx.x;
  const int lane = tid & 31, wv = tid >> 5;
  const int half = lane >> 4, l16 = lane & 15;
  const int b = blockIdx.y;
  const int i0 = blockIdx.x * 64 + wv * 16;
  const int sl = seqlen[b];
  const size_t base = (size_t)b * T_;

  {  // stage this wave's 16 sx rows: 1088 float4, exactly 34 per lane
    float* al = AL + wv * 16 * PXS_;
    for (int u = lane; u < 1088; u += 32) {
      int row = u / 68, c4 = (u % 68) * 4;
      *(float4*)&al[row * PXS_ + c4] =
          *(const float4*)(sx + (base + i0 + row) * (size_t)PXS_ + c4);
    }
  }
  float* wl = WL + wv * 256;
  const float* alw = AL + wv * 16 * PXS_ + l16 * PXS_ + 2 * half;

  v8f acc[8] = {};
  float denom = 0.f;
  const int jtn = (sl + 15) >> 4;  // cols >= sl have zero weight
  const int nc = (jtn + 1) >> 1;   // 32-col chunks
  for (int c = 0; c < nc; ++c) {
    const int jc = c * 32;
    __syncthreads();  // previous chunk fully consumed
    // stage px j-rows: 2176 float4 = 17/thread
    for (int u = tid; u < 2176; u += 128) {
      int row = u / 68, c4 = (u % 68) * 4;
      *(float4*)&BJ[row * PXS_ + c4] =
          *(const float4*)(px + (base + jc + row) * (size_t)PXS_ + c4);
    }
    // stage g1 j-rows k-pair interleaved: 16 pairs x 128, 4 pair-float4/thread
#pragma unroll
    for (int q = 0; q < 4; ++q) {
      int u = tid + 128 * q, kp = u >> 5, n4 = (u & 31) * 4;
      float4 x = *(const float4*)(g1 + (base + jc + 2 * kp) * (size_t)DG_ + n4);
      float4 y =
          *(const float4*)(g1 + (base + jc + 2 * kp + 1) * (size_t)DG_ + n4);
      float4 lo, hi;
      interleave2(x, y, lo, hi);
      *(float4*)&G1L[kp * GPS_ + n4 * 2] = lo;
      *(float4*)&G1L[kp * GPS_ + n4 * 2 + 4] = hi;
    }
    __syncthreads();
#pragma unroll
    for (int sub = 0; sub < 2; ++sub) {
      const int j0 = jc + sub * 16;
      // ---- stage 1: Minkowski Gram tile from LDS (K=260) ----
      const float* bl = BJ + (sub * 16 + l16) * PXS_ + 2 * half;
      v8f s = {};
      for (int k0 = 0; k0 < DHP_; k0 += 4) {
        v2f a = *(const v2f*)(alw + k0);
        v2f bb = *(const v2f*)(bl + k0);
        s = wmma4(a, bb, s);
      }
      // ---- d -> exp(-d) -> threshold -> unnormalized softmax weight ----
      int colv = (j0 + l16) < sl;
#pragma unroll
      for (int v = 0; v < 8; ++v) {
        float xy = fmaxf(s[v], 1.0f);
        float dd = logf(xy + sqrtf(xy * xy - 1.0f + 1e-7f));
        dd = fminf(fmaxf(dd, 1e-6f), 200.0f);
        float ex = __expf(-dd);
        float thr = ex > 0.8f ? ex : 0.0f;
        // logits <= 1 so exp never overflows: no running max needed
        wl[(v + 8 * half) * 16 + l16] = colv ? __expf(thr) : 0.0f;
      }
      // per-wave LDS region, wave-internal DS ops are in-order: no barrier
      if (half == 0) {  // softmax denominator per row
        float rs = 0.f;
#pragma unroll
        for (int cc = 0; cc < 16; ++cc) rs += wl[l16 * 16 + cc];
        denom += rs;
      }
      // ---- stage 2: acc += W(16x16) @ g1[j0:j0+16, :] (LDS-fed) ----
#pragma unroll
      for (int kw = 0; kw < 16; kw += 4) {
        v2f a = *(const v2f*)&wl[l16 * 16 + kw + 2 * half];
        const float* gl =
            &G1L[(sub * 8 + (kw >> 1) + half) * GPS_ + l16 * 2];
#pragma unroll
        for (int t = 0; t < 8; ++t) {
          v2f bb = *(const v2f*)(gl + t * 32);  // one ds_load_b64
          acc[t] = wmma4(a, bb, acc[t]);
        }
      }
    }
  }
  if (half == 0) DL[wv * 16 + l16] = denom;
  __syncthreads();
#pragma unroll
  for (int t = 0; t < 8; ++t) {
#pragma unroll
    for (int v = 0; v < 8; ++v) {
      int m = v + 8 * half;
      int i = i0 + m;
      float val = (i < sl) ? leaky_(acc[t][v] / DL[wv * 16 + m]) : 0.0f;
      outx[(base + i) * 256 + t * 16 + l16] = val;  // x1 -> cols [0,128)
    }
  }
}

// ------- disadj: x2 = leaky(disadj @ g2), LDS table + staged g2 panel -------
__global__ __launch_bounds__(128) void k_disadj(const float* __restrict__ g2,
                                                float* __restrict__ outx) {
  __shared__ __align__(16) float tbl[T_];
  __shared__ __align__(16) float Gsh[2][16 * GPS_];
  const int tid = threadIdx.x;
  const int lane = tid & 31, wv = tid >> 5;
  const int half = lane >> 4, l16 = lane & 15;
  const int b = blockIdx.y;
  const int i0 = blockIdx.x * 64 + wv * 16;
  const float cinv = 0.36787944117144233f;  // 1/e
  for (int d = tid; d < T_; d += 128) tbl[d] = __expf(-cinv * (float)d);

  const size_t base = (size_t)b * T_;
  const float* gp0 = g2 + base * DG_;
  v8f acc[8] = {};
  float4 gx[4], gy[4];
  const int myrow = i0 + l16;

#pragma unroll
  for (int q = 0; q < 4; ++q) {  // preload + stage chunk 0 (k-pair interleave)
    int u = tid + 128 * q, kp = u >> 5, n4 = (u & 31) * 4;
    gx[q] = *(const float4*)(gp0 + (size_t)(2 * kp) * DG_ + n4);
    gy[q] = *(const float4*)(gp0 + (size_t)(2 * kp + 1) * DG_ + n4);
  }
#pragma unroll
  for (int q = 0; q < 4; ++q) {
    int u = tid + 128 * q, kp = u >> 5, n4 = (u & 31) * 4;
    float4 lo, hi;
    interleave2(gx[q], gy[q], lo, hi);
    *(float4*)&Gsh[0][kp * GPS_ + n4 * 2] = lo;
    *(float4*)&Gsh[0][kp * GPS_ + n4 * 2 + 4] = hi;
  }
  __syncthreads();

  const int nchunk = T_ / KC_;  // 64
  for (int c = 0; c < nchunk; ++c) {
    const int cur = c & 1;
    if (c + 1 < nchunk) {  // prefetch next g2 panel (no wait)
      const int k0n = (c + 1) * KC_;
#pragma unroll
      for (int q = 0; q < 4; ++q) {
        int u = tid + 128 * q, kp = u >> 5, n4 = (u & 31) * 4;
        gx[q] = *(const float4*)(gp0 + (size_t)(k0n + 2 * kp) * DG_ + n4);
        gy[q] = *(const float4*)(gp0 + (size_t)(k0n + 2 * kp + 1) * DG_ + n4);
      }
    }
    const int kbase = c * KC_;
#pragma unroll
    for (int ks = 0; ks < 8; ++ks) {  // 64 WMMAs per chunk
      int k = kbase + ks * 4 + 2 * half;
      int d0 = myrow - k;
      d0 = d0 < 0 ? -d0 : d0;
      int d1 = myrow - (k + 1);
      d1 = d1 < 0 ? -d1 : d1;
      v2f a;
      a.x = tbl[d0];
      a.y = tbl[d1];
      const float* gl = &Gsh[cur][(ks * 2 + half) * GPS_ + l16 * 2];
#pragma unroll
      for (int t = 0; t < 8; ++t) {
        v2f bb = *(const v2f*)(gl + t * 32);  // one ds_load_b64
        acc[t] = wmma4(a, bb, acc[t]);
      }
    }
    if (c + 1 < nchunk) {
#pragma unroll
      for (int q = 0; q < 4; ++q) {
        int u = tid + 128 * q, kp = u >> 5, n4 = (u & 31) * 4;
        float4 lo, hi;
        interleave2(gx[q], gy[q], lo, hi);
        *(float4*)&Gsh[cur ^ 1][kp * GPS_ + n4 * 2] = lo;
        *(float4*)&Gsh[cur ^ 1][kp * GPS_ + n4 * 2 + 4] = hi;
      }
    }
    __syncthreads();
  }
#pragma unroll
  for (int t = 0; t < 8; ++t)
#pragma unroll
    for (int v = 0; v < 8; ++v) {
      int i = i0 + v + 8 * half;
      outx[(base + i) * 256 + 128 + t * 16 + l16] = leaky_(acc[t][v]);  // x2
    }
}

// ---------------- classifier head: frame_prob ----------------
__global__ __launch_bounds__(256) void k_frame(const float* __restrict__ outx,
                                               const float* __restrict__ cls_w,
                                               const float* __restrict__ cls_b,
                                               float* __restrict__ frame) {
  int row = blockIdx.x * 8 + (threadIdx.x >> 5);
  int lane = threadIdx.x & 31;
  const float* xr = outx + (size_t)row * 256;
  float s = 0.f;
#pragma unroll
  for (int q = 0; q < 8; ++q) {
    int d = lane + 32 * q;
    float sg = (d == 0) ? -1.f : 1.f;
    s += xr[d] * sg * cls_w[d];
  }
#pragma unroll
  for (int o = 16; o > 0; o >>= 1) s += __shfl_xor(s, o, 32);
  if (lane == 0) frame[row] = 2.0f + 2.0f * s + cls_b[0];
}

// ---------------- top-k mean + sigmoid via in-LDS bitonic sort --------------
__global__ __launch_bounds__(256) void k_clas(const float* __restrict__ frame,
                                              const int* __restrict__ seqlen,
                                              float* __restrict__ mil) {
  __shared__ float vals[T_];
  int b = blockIdx.x, tid = threadIdx.x;
  int sl = seqlen[b];
  for (int t = tid; t < T_; t += 256)
    vals[t] = (t < sl) ? frame[b * T_ + t] : -1e30f;
  __syncthreads();
  for (int k = 2; k <= T_; k <<= 1)
    for (int j = k >> 1; j > 0; j >>= 1) {
      for (int i = tid; i < T_; i += 256) {
        int ixj = i ^ j;
        if (ixj > i) {
          float a = vals[i], bb = vals[ixj];
          bool descRegion = ((i & k) == 0);  // sort descending overall
          if (descRegion ? (a < bb) : (a > bb)) {
            vals[i] = bb;
            vals[ixj] = a;
          }
        }
      }
      __syncthreads();
    }
  if (tid == 0) {
    int kk = sl / 16 + 1;  // <= 129
    float s = 0.f;
    for (int t = 0; t < kk; ++t) s += vals[t];
    float m = s / (float)kk;
    mil[b] = 1.0f / (1.0f + __expf(-m));
  }
}

// ---------------- host launcher ----------------
extern "C" void kernel_launch(void* const* d_in, const int* in_sizes, int n_in,
                              void* d_out, int out_size, void* d_ws,
                              size_t ws_size, hipStream_t stream) {
  const float* inputs = (const float*)d_in[0];
  const int* seqlen   = (const int*)d_in[1];
  const float* w1     = (const float*)d_in[2];
  const float* b1     = (const float*)d_in[3];
  const float* w2     = (const float*)d_in[4];
  const float* b2     = (const float*)d_in[5];
  const float* gw1    = (const float*)d_in[6];
  const float* gw2    = (const float*)d_in[7];
  const float* cls_w  = (const float*)d_in[8];
  const float* cls_b  = (const float*)d_in[9];

  const int M = B_ * T_;  // 16384

  float* ws   = (float*)d_ws;
  float* w1t  = ws;                          // 1024*512
  float* w2t  = w1t + (size_t)DV_ * D1_;     // 512*128
  float* gw1p = w2t + (size_t)D1_ * D2_;     // 260*128
  float* gw2p = gw1p + (size_t)DHP_ * DG_;   // 260*128
  float* xv1  = gw2p + (size_t)DHP_ * DG_;   // 16384*512
  float* xv2  = xv1 + (size_t)M * D1_;       // 16384*128
  float* px   = xv2 + (size_t)M * D2_;       // 16384*272
  float* sx   = px + (size_t)M * PXS_;       // 16384*272
  float* g1   = sx + (size_t)M * PXS_;       // 16384*128
  float* g2   = g1 + (size_t)M * DG_;        // 16384*128
  float* outx = g2 + (size_t)M * DG_;        // 16384*256

  float* mil   = (float*)d_out;  // 8
  float* frame = mil + B_;       // 16384

  // weight prep
  k_transpose<<<dim3((D1_ * DV_ + 255) / 256), 256, 0, stream>>>(w1, w1t, D1_, DV_);
  k_transpose<<<dim3((D2_ * D1_ + 255) / 256), 256, 0, stream>>>(w2, w2t, D2_, D1_);
  k_padcopy<<<dim3((DHP_ * DG_ + 255) / 256), 256, 0, stream>>>(gw1, gw1p, DH_, DG_, DHP_);
  k_padcopy<<<dim3((DHP_ * DG_ + 255) / 256), 256, 0, stream>>>(gw2, gw2p, DH_, DG_, DHP_);

  // MLP
  k_gemm<<<dim3(D1_ / 64, M / 64), 128, 0, stream>>>(inputs, DV_ + DA_, w1t, D1_,
                                                     b1, xv1, D1_, DV_, 1);
  k_gemm<<<dim3(D2_ / 64, M / 64), 128, 0, stream>>>(xv1, D1_, w2t, D2_, b2, xv2,
                                                     D2_, D1_, 1);
  // hyperboloid lift
  k_proj<<<dim3(M / 8), 256, 0, stream>>>(xv2, inputs, px, sx);

  // g1 = proj_x @ gw1, g2 = proj_x @ gw2 (reassociated graph matmuls)
  k_gemm<<<dim3(DG_ / 64, M / 64), 128, 0, stream>>>(px, PXS_, gw1p, DG_, nullptr,
                                                     g1, DG_, DHP_, 0);
  k_gemm<<<dim3(DG_ / 64, M / 64), 128, 0, stream>>>(px, PXS_, gw2p, DG_, nullptr,
                                                     g2, DG_, DHP_, 0);

  // fused attention-style graph aggregations
  const size_t adj_lds =
      (size_t)(4 * 16 * PXS_ + 32 * PXS_ + 16 * GPS_ + 4 * 256 + 64) * sizeof(float);
  k_adjattn<<<dim3(T_ / 64, B_), 128, adj_lds, stream>>>(sx, px, g1, seqlen, outx);
  k_disadj<<<dim3(T_ / 64, B_), 128, 0, stream>>>(g2, outx);

  // head + MIL top-k
  k_frame<<<dim3(M / 8), 256, 0, stream>>>(outx, cls_w, cls_b, frame);
  k_clas<<<dim3(B_), 256, 0, stream>>>(frame, seqlen, mil);
}